// BlockDiagonalAggregator_2190433321665
// MI455X (gfx1250) — compile-verified
//
#include <hip/hip_runtime.h>
#include <stdint.h>

// ---------------------------------------------------------------------------
// BlockDiagonalAggregator on MI455X (gfx1250, wave32)
//   one workgroup per batch b; h[b] (64x512 f32 = 128KB) staged in LDS
//   (320KB/WGP -> 2 blocks per WGP) via GLOBAL_LOAD_ASYNC_TO_LDS_B128
//   (ASYNCcnt-tracked DMA, non-temporal: h is streamed exactly once,
//   512MB => ~22us @ 23.3TB/s). Pooling uses V_WMMA_F32_16X16X4_F32 with
//   alpha broadcast across the M dimension (full f32 precision).
// ---------------------------------------------------------------------------

typedef __attribute__((ext_vector_type(2))) float v2f;
typedef __attribute__((ext_vector_type(8))) float v8f;

#define B_TOT      4096
#define K_SLOTS    64
#define DH         512
#define N_AGENTS   64
#define NEG_INF    (-1e9f)
#define ROW_STRIDE 528   // 512 + 16 pad: lane-halves of WMMA B-fetch hit disjoint LDS bank groups
#define NTHREADS   256   // 8 waves (wave32)

__global__ __launch_bounds__(NTHREADS)
void agg_pool_kernel(const float* __restrict__ h,
                     const float* __restrict__ keys,
                     const int*   __restrict__ sigma,
                     float*       __restrict__ out)
{
    extern __shared__ float smem[];
    float* hS      = smem;                          // 64 x ROW_STRIDE
    float* logitsS = smem + K_SLOTS * ROW_STRIDE;   // 64
    float* alphaS  = logitsS + K_SLOTS;             // 64

    const int b    = blockIdx.x;
    const int tid  = threadIdx.x;
    const int lane = tid & 31;
    const int wave = tid >> 5;

    // ---- Phase A: async-DMA h[b] into LDS (16B/lane, ASYNCcnt-tracked, NT) ----
    // GVS mode: mem_addr = SGPR_base + VGPR_off32; LDS dest = per-lane byte
    // offset within the workgroup LDS allocation (= low 32 bits of the flat
    // address of a __shared__ pointer; LDS aperture lives in the high word).
    {
        const float* hb = h + (size_t)b * (K_SLOTS * DH);
        #pragma unroll
        for (int i = 0; i < (K_SLOTS * DH / 4) / NTHREADS; ++i) {
            const int fidx = (tid + NTHREADS * i) * 4;   // flat float index
            const int row  = fidx >> 9;                  // /512
            const int col  = fidx & 511;
            const uint32_t lds_off = (uint32_t)(uintptr_t)(hS + row * ROW_STRIDE + col);
            const uint32_t goff    = (uint32_t)(fidx * sizeof(float));
            asm volatile("global_load_async_to_lds_b128 %0, %1, %2 th:TH_LOAD_NT"
                         :: "v"(lds_off), "v"(goff), "s"(hb)
                         : "memory");
        }
        asm volatile("s_wait_asynccnt 0x0" ::: "memory");
    }
    __syncthreads();

    // ---- Phase B: logits[k] = valid ? <keys[sigma[k]], h[b,k,:]> : -1e9 ----
    // wave w owns slots w*8 .. w*8+7; 32 lanes reduce over the 512 dims.
    #pragma unroll
    for (int kk = 0; kk < K_SLOTS / 8; ++kk) {
        const int  k     = wave * 8 + kk;
        const int  sig   = sigma[(size_t)b * K_SLOTS + k];
        const bool valid = (unsigned)sig < (unsigned)N_AGENTS;
        const float* wrow = keys + (size_t)(valid ? sig : 0) * DH;  // safe gather (L2-hot)
        float dot = 0.f;
        #pragma unroll
        for (int j = lane; j < DH; j += 32)
            dot = fmaf(wrow[j], hS[k * ROW_STRIDE + j], dot);
        #pragma unroll
        for (int off = 16; off > 0; off >>= 1)
            dot += __shfl_xor(dot, off, 32);
        if (lane == 0) logitsS[k] = valid ? dot : NEG_INF;
    }
    __syncthreads();

    // ---- Softmax over the 64 slots (wave 0 only; 2 values per lane) ----
    if (wave == 0) {
        float l0 = logitsS[lane], l1 = logitsS[lane + 32];
        float m = fmaxf(l0, l1);
        #pragma unroll
        for (int off = 16; off > 0; off >>= 1)
            m = fmaxf(m, __shfl_xor(m, off, 32));
        const float e0 = expf(l0 - m);
        const float e1 = expf(l1 - m);
        float s = e0 + e1;
        #pragma unroll
        for (int off = 16; off > 0; off >>= 1)
            s += __shfl_xor(s, off, 32);
        const float inv = 1.0f / s;
        alphaS[lane]      = e0 * inv;
        alphaS[lane + 32] = e1 * inv;
    }
    __syncthreads();

    // ---- Phase C: out[b,:] = alpha (1x64) x h (64x512) via V_WMMA_F32_16X16X4_F32
    // A(16x4) f32 layout: lanes 0-15 hold K=0 (v0), K=1 (v1); lanes 16-31 hold K=2,K=3.
    // alpha is broadcast over M, so every lane just needs its two K-values.
    // B(4x16) mirrored: v0 = row {0|2} cols 0..15, v1 = row {1|3}.
    // All 16 rows of D are identical; lanes 0-15 (M=0 in c[0]) write the result.
    const int khalf = lane >> 4;     // 0 for lanes 0-15, 1 for lanes 16-31
    const int n     = lane & 15;
    float* outb = out + (size_t)b * DH;
    #pragma unroll
    for (int t = 0; t < 4; ++t) {                  // 4 tiles of 16 cols -> 64 cols per wave
        const int d0 = wave * 64 + t * 16;
        v8f c = {};
        #pragma unroll
        for (int kc = 0; kc < K_SLOTS; kc += 4) {
            v2f a, bm;
            a.x  = alphaS[kc + khalf * 2 + 0];
            a.y  = alphaS[kc + khalf * 2 + 1];
            bm.x = hS[(kc + khalf * 2 + 0) * ROW_STRIDE + d0 + n];
            bm.y = hS[(kc + khalf * 2 + 1) * ROW_STRIDE + d0 + n];
            c = __builtin_amdgcn_wmma_f32_16x16x4_f32(false, a, false, bm,
                                                      (short)0, c, false, false);
        }
        if (lane < 16) outb[d0 + lane] = c[0];
    }
}

extern "C" void kernel_launch(void* const* d_in, const int* in_sizes, int n_in,
                              void* d_out, int out_size, void* d_ws, size_t ws_size,
                              hipStream_t stream)
{
    (void)in_sizes; (void)n_in; (void)d_ws; (void)ws_size; (void)out_size;
    const float* h     = (const float*)d_in[0];   // (B, K, D_H) f32
    const float* keys  = (const float*)d_in[1];   // (N_AGENTS, D_H) f32
    const int*   sigma = (const int*)d_in[2];     // (B, K) int32
    float*       out   = (float*)d_out;           // (B, D_H) f32

    const size_t smem_bytes = (size_t)(K_SLOTS * ROW_STRIDE + 2 * K_SLOTS) * sizeof(float);
    agg_pool_kernel<<<B_TOT, NTHREADS, smem_bytes, stream>>>(h, keys, sigma, out);
}